// ProjectionPursuitProbe_77592879169970
// MI455X (gfx1250) — compile-verified
//
#include <hip/hip_runtime.h>
#include <hip/hip_bf16.h>
#include <math.h>

// Problem constants (from reference)
#define BATCH 8
#define SEQ   4096
#define FDIM  1024
#define RANK  64
#define NTOK  (BATCH * SEQ)       // 32768 rows
#define BM    64                  // rows per workgroup tile
#define HP    1032                // padded LDS row: 1024 + 4 pads x 2 DWORDs (TDM pad_enable)
#define CP    68                  // padded C row (64 + 4)

typedef float v2f __attribute__((ext_vector_type(2)));
typedef float v8f __attribute__((ext_vector_type(8)));
typedef unsigned int v4u __attribute__((ext_vector_type(4)));
typedef int v8i __attribute__((ext_vector_type(8)));
typedef int v4i __attribute__((ext_vector_type(4)));

// LDS offset (in floats) of element (r, f) of the H tile with TDM intra-row
// padding: 2 DWORDs inserted after every 256 DWORDs of the DMA stream.
__device__ __forceinline__ int hsIdx(int r, int f) {
    return r * HP + f + ((f >> 8) << 1);
}

// -------------------------------------------------------------------------
// Kernel 1: Gram-Schmidt on probe (1024 x 64), single workgroup.
// Emits WMMA-ready operand layouts into workspace:
//   Utp[((r>>1)*1024 + f)*2 + (r&1)] = U[f][r]        (B-matrix for phase B)
//   Vp [((f>>1)*64   + r)*2 + (f&1)] = U[f][r]/sq[r]  (B-matrix for phase A)
// -------------------------------------------------------------------------
__global__ void pp_gram_schmidt(const float* __restrict__ probe,
                                float* __restrict__ Utp,
                                float* __restrict__ Vp) {
    extern __shared__ float smem[];
    float* Us    = smem;                 // [64][1024], row r contiguous in f
    float* vcol  = Us + RANK * FDIM;     // [1024]
    float* coeff = vcol + FDIM;          // [64]
    float* sq    = coeff + RANK;         // [64]
    float* part  = sq + RANK;            // [256]

    const int t = threadIdx.x;
    const int lane = t & 31;
    const int wave = t >> 5;

    // column 0
    float psq = 0.0f;
    for (int f = t; f < FDIM; f += 256) {
        float v = probe[f * RANK + 0];
        Us[f] = v;
        psq += v * v;
    }
    part[t] = psq;
    __syncthreads();
    if (t == 0) {
        float s = 0.0f;
        for (int k = 0; k < 256; ++k) s += part[k];
        sq[0] = s;
    }
    __syncthreads();

    for (int i = 1; i < RANK; ++i) {
        for (int f = t; f < FDIM; f += 256) vcol[f] = probe[f * RANK + i];
        __syncthreads();

        // dot(U_j, v) for j < i : wave w handles j = w, w+8, ...
        for (int j = wave; j < i; j += 8) {
            const float* Uj = Us + j * FDIM;
            float d = 0.0f;
            for (int f = lane; f < FDIM; f += 32) d += Uj[f] * vcol[f];
            d += __shfl_xor(d, 16, 32);
            d += __shfl_xor(d, 8, 32);
            d += __shfl_xor(d, 4, 32);
            d += __shfl_xor(d, 2, 32);
            d += __shfl_xor(d, 1, 32);
            if (lane == 0) coeff[j] = d / sq[j];
        }
        __syncthreads();

        float ps = 0.0f;
        for (int f = t; f < FDIM; f += 256) {
            float u = vcol[f];
            for (int j = 0; j < i; ++j) u -= coeff[j] * Us[j * FDIM + f];
            Us[i * FDIM + f] = u;
            ps += u * u;
        }
        part[t] = ps;
        __syncthreads();
        if (t == 0) {
            float s = 0.0f;
            for (int k = 0; k < 256; ++k) s += part[k];
            sq[i] = s;
        }
        __syncthreads();
    }

    // emit swizzled operands
    for (int idx = t; idx < RANK * FDIM; idx += 256) {
        int r = idx >> 10;       // rank
        int f = idx & 1023;      // feature
        float u = Us[r * FDIM + f];
        Utp[((r >> 1) * FDIM + f) * 2 + (r & 1)] = u;
        Vp[((f >> 1) * RANK + r) * 2 + (f & 1)]  = u / sq[r];
    }
}

// -------------------------------------------------------------------------
// Kernel 2: fused projection + renorm over a 64-token tile.
//   Stage:   TDM tensor_load_to_lds pulls the 64x1024 f32 tile into LDS,
//            inserting 2 pad DWORDs per 256 (bank-conflict-free WMMA reads).
//   Phase A: C[64x64]  = Htile[64x1024] * Vp          (v_wmma_f32_16x16x4_f32)
//   Phase B: Htile    -= C * Ut                       (v_wmma_f32_16x16x4_f32)
//   Rescale rows by old_norm/new_norm; row s==0 restored to original.
// 256 threads = 8 waves; wave w owns row-block mi = w>>1 (16 rows).
// -------------------------------------------------------------------------
__global__ void pp_project(const float* __restrict__ H,
                           const float* __restrict__ Utp,
                           const float* __restrict__ Vp,
                           float* __restrict__ out) {
    extern __shared__ float smem[];
    float* Hs      = smem;                 // [64][HP]  (TDM-padded)
    float* Cs      = Hs + BM * HP;         // [64][CP]
    float* rowOld  = Cs + BM * CP;         // [64]
    float* scaleS  = rowOld + BM;          // [64]
    float* oldPart = scaleS + BM;          // [64][4]
    float* newPart = oldPart + BM * 4;     // [64][2]

    const int t = threadIdx.x;
    const int lane = t & 31;
    const int wave = t >> 5;
    const int laneLo = lane & 15;
    const int laneHi = lane >> 4;          // 0 for lanes 0-15, 1 for 16-31

    const int rowBase = (int)blockIdx.x * BM;
    const float* Hblk = H + (size_t)rowBase * FDIM;
    float* outBlk = out + (size_t)rowBase * FDIM;

    // ---- stage H tile into LDS via Tensor Data Mover (wave 0 issues) ----
    if (wave == 0) {
        unsigned long long ga = (unsigned long long)(uintptr_t)Hblk;
        unsigned int ldsa = (unsigned int)(uintptr_t)Hs;   // LDS aperture low bits
        v4u g0;
        g0.x = 1u;                                          // count=1, flags 0
        g0.y = ldsa;                                        // lds_addr (bytes)
        g0.z = (unsigned int)(ga & 0xFFFFFFFFu);            // global_addr[31:0]
        g0.w = (unsigned int)((ga >> 32) & 0x01FFFFFFu)     // global_addr[56:32]
               | (2u << 30);                                // type = 2 (image)
        v8i g1;
        g1[0] = (int)((2u << 16)      // data_size = 4 bytes
                    | (1u << 20)      // pad_enable
                    | (7u << 22)      // pad_interval: 256 DWORDs
                    | (1u << 25));    // pad_amount: 2 DWORDs
        g1[1] = (int)(1024u << 16);   // tensor_dim0 = 1024 (lo16 in [31:16])
        g1[2] = (int)(64u << 16);     // tensor_dim0 hi = 0 | tensor_dim1 = 64
        g1[3] = (int)(1024u << 16);   // tensor_dim1 hi = 0 | tile_dim0 = 1024
        g1[4] = 64;                   // tile_dim1 = 64, tile_dim2 = 0
        g1[5] = 1024;                 // tensor_dim0_stride = 1024 elements
        g1[6] = 0;
        g1[7] = 0;
        v4i gz4 = {0, 0, 0, 0};
        v8i gz8 = {0, 0, 0, 0, 0, 0, 0, 0};
        // 6-arg form (clang-23 / therock-10.0 ABI)
        __builtin_amdgcn_tensor_load_to_lds(g0, g1, gz4, gz4, gz8, 0);
        __builtin_amdgcn_s_wait_tensorcnt(0);
    }
    // warm L2 for the phase-B operand while the DMA is in flight
    for (int off = t * 32; off < RANK * FDIM; off += 256 * 32)
        __builtin_prefetch(Utp + off, 0, 3);
    __syncthreads();

    // ---- old row norms^2 (deterministic partials) ----
    {
        int r = t >> 2, q = t & 3;
        const float* hr = Hs + r * HP + q * 256 + q * 2;    // chunk is contiguous
        float s = 0.0f;
        for (int j = 0; j < 256; ++j) s += hr[j] * hr[j];
        oldPart[r * 4 + q] = s;
    }

    // ---- Phase A: C = Htile * V   (each wave: 2 output tiles, same mi) ----
    const int mi = wave >> 1;
    const int nbase = (wave & 1) * 2;        // column tiles nbase, nbase+1
    v8f c0 = {};
    v8f c1 = {};
    {
        const int arow = mi * 16 + laneLo;
        const int n0 = nbase * 16 + laneLo;
        const int n1 = n0 + 16;
        const float* hrow = Hs + arow * HP;
#pragma unroll 4
        for (int k = 0; k < FDIM; k += 4) {
            const int k0 = k + laneHi * 2;
            v2f a  = *(const v2f*)(hrow + k0 + ((k0 >> 8) << 1));
            v2f b0 = *(const v2f*)(Vp + ((k0 >> 1) * RANK + n0) * 2);
            v2f b1 = *(const v2f*)(Vp + ((k0 >> 1) * RANK + n1) * 2);
            c0 = __builtin_amdgcn_wmma_f32_16x16x4_f32(false, a, false, b0,
                                                       (short)0, c0, false, false);
            c1 = __builtin_amdgcn_wmma_f32_16x16x4_f32(false, a, false, b1,
                                                       (short)0, c1, false, false);
        }
    }
    // dump C to LDS (D layout: VGPR g -> row m0+g (+8 for hi half), col = lane&15)
    {
        const int m0 = mi * 16 + laneHi * 8;
        const int nA = nbase * 16 + laneLo;
#pragma unroll
        for (int g = 0; g < 8; ++g) {
            Cs[(m0 + g) * CP + nA]      = c0[g];
            Cs[(m0 + g) * CP + nA + 16] = c1[g];
        }
    }
    __syncthreads();

    if (t < BM)
        rowOld[t] = oldPart[t * 4] + oldPart[t * 4 + 1] +
                    oldPart[t * 4 + 2] + oldPart[t * 4 + 3];

    // ---- Phase B: Htile -= C * Ut ; accumulate new norms ----
    float accs[8] = {0, 0, 0, 0, 0, 0, 0, 0};
    const int m0 = mi * 16;
    for (int tt = 0; tt < 32; ++tt) {
        const int nf = (wave & 1) + tt * 2;          // this wave's column tile
        const int fcol = nf * 16 + laneLo;
        const int foff = fcol + ((fcol >> 8) << 1);  // padded column offset
        const int arow = m0 + laneLo;
        v8f p = {};
#pragma unroll
        for (int r = 0; r < RANK; r += 4) {
            const int r0 = r + laneHi * 2;
            v2f a = *(const v2f*)(Cs + arow * CP + r0);
            v2f b = *(const v2f*)(Utp + ((r0 >> 1) * FDIM + fcol) * 2);
            p = __builtin_amdgcn_wmma_f32_16x16x4_f32(false, a, false, b,
                                                      (short)0, p, false, false);
        }
        const int mrow = m0 + laneHi * 8;
#pragma unroll
        for (int g = 0; g < 8; ++g) {
            float* hp = Hs + (mrow + g) * HP + foff;
            float hv = *hp - p[g];
            *hp = hv;
            accs[g] += hv * hv;
        }
    }
    // reduce accs within each 16-lane half (wave32): each half is one row set
#pragma unroll
    for (int g = 0; g < 8; ++g) {
        float v = accs[g];
        v += __shfl_xor(v, 1, 32);
        v += __shfl_xor(v, 2, 32);
        v += __shfl_xor(v, 4, 32);
        v += __shfl_xor(v, 8, 32);
        if (laneLo == 0)
            newPart[(m0 + laneHi * 8 + g) * 2 + (wave & 1)] = v;
    }
    __syncthreads();

    if (t < BM)
        scaleS[t] = sqrtf(rowOld[t] / (newPart[t * 2] + newPart[t * 2 + 1]));
    __syncthreads();

    // ---- final store (token s==0 keeps original hidden state) ----
    // float2 granularity keeps 8B alignment across the intra-row pads.
    for (int idx = t; idx < BM * (FDIM / 2); idx += 256) {
        int r = idx >> 9;
        int c2 = idx & 511;
        int f = c2 * 2;
        float sc = scaleS[r];
        const float* src = Hs + r * HP + f + ((f >> 8) << 1);
        float2 v;
        v.x = src[0] * sc;
        v.y = src[1] * sc;
        if (((rowBase + r) & (SEQ - 1)) == 0)
            v = ((const float2*)Hblk)[r * 512 + c2];
        ((float2*)outBlk)[r * 512 + c2] = v;
    }
}

// -------------------------------------------------------------------------
extern "C" void kernel_launch(void* const* d_in, const int* in_sizes, int n_in,
                              void* d_out, int out_size, void* d_ws, size_t ws_size,
                              hipStream_t stream) {
    const float* H = (const float*)d_in[0];       // (8, 4096, 1024) fp32
    const float* probe = (const float*)d_in[1];   // (1024, 64) fp32
    float* out = (float*)d_out;

    float* Utp = (float*)d_ws;                    // 64*1024 floats (256 KB)
    float* Vp  = Utp + RANK * FDIM;               // 64*1024 floats (256 KB)

    // Gram-Schmidt: one workgroup, 256 threads.
    size_t gsLds = (size_t)(RANK * FDIM + FDIM + RANK + RANK + 256) * sizeof(float);
    pp_gram_schmidt<<<1, 256, gsLds, stream>>>(probe, Utp, Vp);

    // Fused projection: 512 workgroups x 256 threads, 64-token tiles.
    size_t mainLds = (size_t)(BM * HP + BM * CP + BM + BM + BM * 4 + BM * 2) * sizeof(float);
    pp_project<<<NTOK / BM, 256, mainLds, stream>>>(H, Utp, Vp, out);
}